// GCM_41085657153564
// MI455X (gfx1250) — compile-verified
//
#include <hip/hip_runtime.h>
#include <hip/hip_bf16.h>

#define N_NODES 2048
#define B_GRAPH 32
#define NPG     64
#define E_EDGES 32768
#define D_DIM   128
#define H_HEADS 4
#define DH_DIM  32
#define L_LAYERS 4

typedef __attribute__((ext_vector_type(2))) float v2f;
typedef __attribute__((ext_vector_type(8))) float v8f;

#if __has_builtin(__builtin_amdgcn_tensor_load_to_lds) && \
    __has_builtin(__builtin_amdgcn_s_wait_tensorcnt)
#define HAVE_TDM 1
typedef unsigned int u32x4 __attribute__((ext_vector_type(4)));
typedef int          i32x8 __attribute__((ext_vector_type(8)));
typedef int          i32x4 __attribute__((ext_vector_type(4)));
#else
#define HAVE_TDM 0
#endif

// ---------------------------------------------------------------------------
// Generic fp32 WMMA GEMM:  C[M,Nout] = A[M,K] @ B + bias
//   TRANSB==0: B[k, col] = W[k*Nout + col]   (row-major K x Nout)
//   TRANSB==1: B[k, col] = W[col*K + k]      (i.e. A @ W.T, W is Nout x K)
// grid = (M/16, Nout/64), block = 128 threads (4 waves, one 16x16 tile each).
// A tile (16 x K fp32) staged into LDS by the Tensor Data Mover (one
// TENSOR_LOAD_TO_LDS per workgroup, issued by wave 0; TDM ignores EXEC),
// fenced with s_wait_tensorcnt + workgroup barrier.
// ---------------------------------------------------------------------------
template <int TRANSB, int K>
__global__ __launch_bounds__(128) void gemm_f32_wmma(
    const float* __restrict__ A, const float* __restrict__ W,
    const float* __restrict__ bias, float* __restrict__ C,
    int M, int Nout)
{
    __shared__ float As[16 * K];
    const int tid  = threadIdx.x;
    const int wave = tid >> 5;
    const int lane = tid & 31;
    const int half = lane >> 4;
    const int l16  = lane & 15;
    const int row0 = blockIdx.x * 16;
    const int col0 = blockIdx.y * 64 + wave * 16;
    const int col  = col0 + l16;

#if HAVE_TDM
    if (wave == 0) {
        // Build D# for a 16-row x K-col fp32 tile, contiguous rows (stride K).
        unsigned long long ga = (unsigned long long)(const void*)(A + (size_t)row0 * K);
        unsigned int lds = (unsigned int)(unsigned long long)(const void*)As; // LDS aperture offset
        u32x4 g0;
        g0[0] = 1u;                                          // count=1 (valid user D#)
        g0[1] = lds;                                         // lds_addr
        g0[2] = (unsigned int)ga;                            // global_addr[31:0]
        g0[3] = (unsigned int)((ga >> 32) & 0x01FFFFFFu)     // global_addr[56:32]
              | (2u << 30);                                  // type=2 ("image")
        i32x8 g1;
        g1[0] = (int)(2u << 16);                             // data_size=2 (4 bytes)
        g1[1] = (int)(((unsigned)K & 0xFFFFu) << 16);        // tensor_dim0[15:0]
        g1[2] = (int)(((unsigned)K >> 16) | ((unsigned)(N_NODES & 0xFFFF) << 16)); // td0 hi | td1 lo
        g1[3] = (int)(((unsigned)N_NODES >> 16) | ((unsigned)K << 16));            // td1 hi | tile_dim0=K
        g1[4] = 16;                                          // tile_dim1=16, tile_dim2=0
        g1[5] = K;                                           // tensor_dim0_stride[31:0] = K
        g1[6] = 0;
        g1[7] = 0;
        i32x4 z4 = {0, 0, 0, 0};
        i32x8 z8 = {0, 0, 0, 0, 0, 0, 0, 0};
        __builtin_amdgcn_tensor_load_to_lds(g0, g1, z4, z4, z8, 0);
        __builtin_amdgcn_s_wait_tensorcnt((short)0);
    }
    __syncthreads();
#else
    for (int i = tid; i < 16 * K; i += 128)
        As[i] = A[(size_t)(row0 + i / K) * K + (i % K)];
    __syncthreads();
#endif

    // keep the shared weight panel hot in cache
    if (TRANSB) __builtin_prefetch(W + (size_t)col * K, 0, 0);
    else        __builtin_prefetch(W + col, 0, 0);

    const float2* As2 = (const float2*)As;
    v8f acc = {};
    for (int k0 = 0; k0 < K; k0 += 4) {
        const int kk = k0 + 2 * half;
        float2 af = As2[(l16 * K + k0) / 2 + half];   // A[row, kk], A[row, kk+1]
        v2f a; a.x = af.x; a.y = af.y;
        v2f b;
        if (TRANSB) {
            float2 bf = *(const float2*)(W + (size_t)col * K + kk);
            b.x = bf.x; b.y = bf.y;
        } else {
            b.x = W[(size_t)kk * Nout + col];
            b.y = W[(size_t)(kk + 1) * Nout + col];
        }
        acc = __builtin_amdgcn_wmma_f32_16x16x4_f32(false, a, false, b,
                                                    (short)0, acc, false, false);
    }
    const float bv = bias ? bias[col] : 0.0f;
    for (int r = 0; r < 8; ++r)
        C[(size_t)(row0 + r + 8 * half) * Nout + col] = acc[r] + bv;
}

// ---------------------------------------------------------------------------
// GENConv pieces
// ---------------------------------------------------------------------------
__global__ void zero_f32(float* p, int n)
{
    int i = blockIdx.x * blockDim.x + threadIdx.x;
    if (i < n) p[i] = 0.0f;
}

// one block per edge, 128 threads = feature dims
__global__ __launch_bounds__(128) void edge_agg(
    const float* __restrict__ x, const int* __restrict__ ei,
    const float* __restrict__ ea, float* __restrict__ agg,
    float* __restrict__ cnt)
{
    const int e = blockIdx.x;
    const int d = threadIdx.x;
    const int src = ei[e];
    const int dst = ei[E_EDGES + e];
    float m = fmaxf(x[(size_t)src * D_DIM + d] + ea[(size_t)e * D_DIM + d], 0.0f) + 1e-7f;
    atomicAdd(&agg[(size_t)dst * D_DIM + d], m);
    if (d == 0) atomicAdd(&cnt[dst], 1.0f);
}

__global__ void finish_agg(float* __restrict__ agg, const float* __restrict__ cnt,
                           const float* __restrict__ x)
{
    int i = blockIdx.x * blockDim.x + threadIdx.x;
    if (i < N_NODES * D_DIM) {
        int n = i / D_DIM;
        agg[i] = agg[i] / fmaxf(cnt[n], 1.0f) + x[i];
    }
}

// one block per column of h1[N,256]; biased var like torch BN (training)
__global__ __launch_bounds__(256) void bn_stats(
    const float* __restrict__ h1, const float* __restrict__ g,
    const float* __restrict__ be, float* __restrict__ scl, float* __restrict__ shf)
{
    __shared__ float s1[256], s2[256];
    const int col = blockIdx.x;
    const int t = threadIdx.x;
    float a = 0.0f, b = 0.0f;
    for (int n = t; n < N_NODES; n += 256) {
        float v = h1[(size_t)n * 256 + col];
        a += v; b += v * v;
    }
    s1[t] = a; s2[t] = b; __syncthreads();
    for (int off = 128; off > 0; off >>= 1) {
        if (t < off) { s1[t] += s1[t + off]; s2[t] += s2[t + off]; }
        __syncthreads();
    }
    if (t == 0) {
        float mean = s1[0] * (1.0f / N_NODES);
        float var  = s2[0] * (1.0f / N_NODES) - mean * mean;
        float inv  = rsqrtf(var + 1e-5f);
        float sc   = g[col] * inv;
        scl[col] = sc;
        shf[col] = be[col] - mean * sc;
    }
}

__global__ void bn_relu(float* __restrict__ h1, const float* __restrict__ scl,
                        const float* __restrict__ shf)
{
    int i = blockIdx.x * blockDim.x + threadIdx.x;
    if (i < N_NODES * 256) {
        int c = i & 255;
        h1[i] = fmaxf(h1[i] * scl[c] + shf[c], 0.0f);
    }
}

// ---------------------------------------------------------------------------
// Block-diagonal cross attention: block = one graph (64 q x 64 kv),
// wave = one head (DH=32). qkv buffers are [N,384] = q|k|v.
// ---------------------------------------------------------------------------
__global__ __launch_bounds__(128) void attn_wmma(
    const float* __restrict__ qkv_q, const float* __restrict__ qkv_kv,
    float* __restrict__ out)
{
    __shared__ float sc[H_HEADS][64][64];   // 64 KB
    const int g    = blockIdx.x;
    const int tid  = threadIdx.x;
    const int h    = tid >> 5;
    const int lane = tid & 31;
    const int half = lane >> 4;
    const int l16  = lane & 15;
    const int base = g * NPG;

    // S = (Q @ K^T) / sqrt(DH)
    for (int mt = 0; mt < 4; ++mt) {
        for (int nt = 0; nt < 4; ++nt) {
            v8f acc = {};
            for (int k0 = 0; k0 < DH_DIM; k0 += 4) {
                const int kk = k0 + 2 * half;
                float2 aq = *(const float2*)(qkv_q  + (size_t)(base + mt * 16 + l16) * 384 + h * 32 + kk);
                float2 bk = *(const float2*)(qkv_kv + (size_t)(base + nt * 16 + l16) * 384 + 128 + h * 32 + kk);
                v2f a; a.x = aq.x; a.y = aq.y;
                v2f b; b.x = bk.x; b.y = bk.y;
                acc = __builtin_amdgcn_wmma_f32_16x16x4_f32(false, a, false, b,
                                                            (short)0, acc, false, false);
            }
            for (int r = 0; r < 8; ++r)
                sc[h][mt * 16 + r + 8 * half][nt * 16 + l16] = acc[r] * 0.1767766952966369f;
        }
    }
    // per-row softmax (uniform control flow: 2 rows per lane)
    for (int rr = lane; rr < 64; rr += 32) {
        float mx = -1e30f;
        for (int j = 0; j < 64; ++j) mx = fmaxf(mx, sc[h][rr][j]);
        float s = 0.0f;
        for (int j = 0; j < 64; ++j) { float e = __expf(sc[h][rr][j] - mx); sc[h][rr][j] = e; s += e; }
        float inv = 1.0f / s;
        for (int j = 0; j < 64; ++j) sc[h][rr][j] *= inv;
    }
    // O = P @ V
    for (int mt = 0; mt < 4; ++mt) {
        for (int nt = 0; nt < 2; ++nt) {
            v8f acc = {};
            for (int k0 = 0; k0 < 64; k0 += 4) {
                const int kk = k0 + 2 * half;
                float2 ap = *(const float2*)(&sc[h][mt * 16 + l16][kk]);
                v2f a; a.x = ap.x; a.y = ap.y;
                v2f b;
                b.x = qkv_kv[(size_t)(base + kk)     * 384 + 256 + h * 32 + nt * 16 + l16];
                b.y = qkv_kv[(size_t)(base + kk + 1) * 384 + 256 + h * 32 + nt * 16 + l16];
                acc = __builtin_amdgcn_wmma_f32_16x16x4_f32(false, a, false, b,
                                                            (short)0, acc, false, false);
            }
            for (int r = 0; r < 8; ++r)
                out[(size_t)(base + mt * 16 + r + 8 * half) * D_DIM + h * 32 + nt * 16 + l16] = acc[r];
        }
    }
}

// ---------------------------------------------------------------------------
// LayerNorm over last dim (128): one block per node
// ---------------------------------------------------------------------------
__global__ __launch_bounds__(128) void layernorm128(
    const float* __restrict__ in, const float* __restrict__ g,
    const float* __restrict__ b, float* __restrict__ outp)
{
    __shared__ float red[128];
    const int n = blockIdx.x, t = threadIdx.x;
    float v = in[(size_t)n * D_DIM + t];
    red[t] = v; __syncthreads();
    for (int off = 64; off > 0; off >>= 1) { if (t < off) red[t] += red[t + off]; __syncthreads(); }
    float mu = red[0] * (1.0f / D_DIM);
    __syncthreads();
    float d = v - mu;
    red[t] = d * d; __syncthreads();
    for (int off = 64; off > 0; off >>= 1) { if (t < off) red[t] += red[t + off]; __syncthreads(); }
    float var = red[0] * (1.0f / D_DIM);
    outp[(size_t)n * D_DIM + t] = d * rsqrtf(var + 1e-5f) * g[t] + b[t];
}

// ---------------------------------------------------------------------------
// Pooling + classifier
// ---------------------------------------------------------------------------
__global__ __launch_bounds__(128) void gate_kernel(
    const float* __restrict__ x, const float* __restrict__ wg,
    const float* __restrict__ bg, float* __restrict__ gate)
{
    __shared__ float red[128];
    const int n = blockIdx.x, t = threadIdx.x;
    red[t] = x[(size_t)n * D_DIM + t] * wg[t];
    __syncthreads();
    for (int off = 64; off > 0; off >>= 1) { if (t < off) red[t] += red[t + off]; __syncthreads(); }
    if (t == 0) gate[n] = 1.0f / (1.0f + __expf(-(red[0] + bg[0])));
}

__global__ __launch_bounds__(128) void pool_kernel(
    const float* __restrict__ x, const float* __restrict__ gate,
    float* __restrict__ pooled, int off)
{
    __shared__ float w[64];
    __shared__ float red[64];
    const int g = blockIdx.x, t = threadIdx.x;
    if (t < 64) { w[t] = gate[g * NPG + t]; red[t] = w[t]; }
    __syncthreads();
    for (int o = 32; o > 0; o >>= 1) { if (t < o) red[t] = fmaxf(red[t], red[t + o]); __syncthreads(); }
    float mx = red[0];
    __syncthreads();
    if (t < 64) { w[t] = __expf(w[t] - mx); red[t] = w[t]; }
    __syncthreads();
    for (int o = 32; o > 0; o >>= 1) { if (t < o) red[t] += red[t + o]; __syncthreads(); }
    float inv = 1.0f / (red[0] + 1e-16f);
    __syncthreads();
    float acc = 0.0f;
    for (int m = 0; m < NPG; ++m)
        acc += w[m] * inv * x[(size_t)(g * NPG + m) * D_DIM + t];
    pooled[(size_t)g * 256 + off + t] = acc;
}

__global__ void classify(const float* __restrict__ pooled, const float* __restrict__ Wc,
                         const float* __restrict__ bc, float* __restrict__ out)
{
    const int b = threadIdx.x;
    if (b >= B_GRAPH) return;
    float l0 = bc[0], l1 = bc[1];
    for (int j = 0; j < 256; ++j) {
        float p = pooled[(size_t)b * 256 + j];
        l0 += p * Wc[j * 2];
        l1 += p * Wc[j * 2 + 1];
    }
    float m = fmaxf(l0, l1);
    float e0 = __expf(l0 - m), e1 = __expf(l1 - m);
    float s = e0 + e1;
    out[b * 2]     = e0 / s;
    out[b * 2 + 1] = e1 / s;
}

// ---------------------------------------------------------------------------
// Host orchestration
// ---------------------------------------------------------------------------
extern "C" void kernel_launch(void* const* d_in, const int* in_sizes, int n_in,
                              void* d_out, int out_size, void* d_ws, size_t ws_size,
                              hipStream_t stream)
{
    const float* in_xs = (const float*)d_in[0];
    const float* in_xt = (const float*)d_in[1];
    const float* ea_s  = (const float*)d_in[2];
    const float* ea_t  = (const float*)d_in[3];
    const float* W1    = (const float*)d_in[4];
    const float* b1    = (const float*)d_in[5];
    const float* bng   = (const float*)d_in[6];
    const float* bnb   = (const float*)d_in[7];
    const float* W2    = (const float*)d_in[8];
    const float* b2    = (const float*)d_in[9];
    const float* ipw   = (const float*)d_in[10];
    const float* ipb   = (const float*)d_in[11];
    const float* opw   = (const float*)d_in[12];
    const float* opb   = (const float*)d_in[13];
    const float* lng   = (const float*)d_in[14];
    const float* lnb   = (const float*)d_in[15];
    const float* Wg    = (const float*)d_in[16];
    const float* bg    = (const float*)d_in[17];
    const float* Wc    = (const float*)d_in[18];
    const float* bc    = (const float*)d_in[19];
    const int*   ei_s  = (const int*)d_in[20];
    const int*   ei_t  = (const int*)d_in[21];

    const int ND = N_NODES * D_DIM;
    float* ws   = (float*)d_ws;
    float* xs   = ws;                  // N*D
    float* xt   = xs + ND;             // N*D
    float* agg  = xt + ND;             // N*D
    float* cnt  = agg + ND;            // N  (contiguous after agg -> zero together)
    float* h1   = cnt + N_NODES;       // N*256
    float* qkvs = h1 + N_NODES * 256;  // N*384
    float* qkvt = qkvs + N_NODES * 384;// N*384
    float* o_s  = qkvt + N_NODES * 384;// N*D
    float* o_t  = o_s + ND;            // N*D
    float* tmp  = o_t + ND;            // N*D
    float* scl  = tmp + ND;            // 256
    float* shf  = scl + 256;           // 256
    float* gat  = shf + 256;           // N
    float* pooled = gat + N_NODES;     // B*256

    (void)hipMemcpyAsync(xs, in_xs, (size_t)ND * sizeof(float), hipMemcpyDeviceToDevice, stream);
    (void)hipMemcpyAsync(xt, in_xt, (size_t)ND * sizeof(float), hipMemcpyDeviceToDevice, stream);

    const dim3 blk128(128);
    for (int i = 0; i < L_LAYERS; ++i) {
        const float* W1i = W1 + (size_t)i * D_DIM * 256;
        const float* b1i = b1 + (size_t)i * 256;
        const float* gni = bng + (size_t)i * 256;
        const float* bni = bnb + (size_t)i * 256;
        const float* W2i = W2 + (size_t)i * 256 * D_DIM;
        const float* b2i = b2 + (size_t)i * D_DIM;
        const float* ipwi = ipw + (size_t)i * 384 * D_DIM;
        const float* ipbi = ipb + (size_t)i * 384;
        const float* opwi = opw + (size_t)i * D_DIM * D_DIM;
        const float* opbi = opb + (size_t)i * D_DIM;
        const float* lngi = lng + (size_t)i * D_DIM;
        const float* lnbi = lnb + (size_t)i * D_DIM;

        // ---- GENConv for each side ----
        for (int side = 0; side < 2; ++side) {
            float* x = side ? xt : xs;
            const int* ei = side ? ei_t : ei_s;
            const float* ea = side ? ea_t : ea_s;
            zero_f32<<<(ND + N_NODES + 255) / 256, 256, 0, stream>>>(agg, ND + N_NODES);
            edge_agg<<<E_EDGES, blk128, 0, stream>>>(x, ei, ea, agg, cnt);
            finish_agg<<<(ND + 255) / 256, 256, 0, stream>>>(agg, cnt, x);
            gemm_f32_wmma<0, 128><<<dim3(N_NODES / 16, 256 / 64), blk128, 0, stream>>>(
                agg, W1i, b1i, h1, N_NODES, 256);
            bn_stats<<<256, 256, 0, stream>>>(h1, gni, bni, scl, shf);
            bn_relu<<<(N_NODES * 256 + 255) / 256, 256, 0, stream>>>(h1, scl, shf);
            gemm_f32_wmma<0, 256><<<dim3(N_NODES / 16, 128 / 64), blk128, 0, stream>>>(
                h1, W2i, b2i, x, N_NODES, 128);
        }

        // ---- fused QKV projections (x @ ipw.T), one GEMM per side ----
        gemm_f32_wmma<1, 128><<<dim3(N_NODES / 16, 384 / 64), blk128, 0, stream>>>(
            xs, ipwi, ipbi, qkvs, N_NODES, 384);
        gemm_f32_wmma<1, 128><<<dim3(N_NODES / 16, 384 / 64), blk128, 0, stream>>>(
            xt, ipwi, ipbi, qkvt, N_NODES, 384);

        // ---- block-diagonal cross attention ----
        attn_wmma<<<B_GRAPH, blk128, 0, stream>>>(qkvs, qkvt, o_s);
        attn_wmma<<<B_GRAPH, blk128, 0, stream>>>(qkvt, qkvs, o_t);

        // ---- out projection (o @ opw.T) + LayerNorm -> new xs/xt ----
        gemm_f32_wmma<1, 128><<<dim3(N_NODES / 16, 128 / 64), blk128, 0, stream>>>(
            o_s, opwi, opbi, tmp, N_NODES, 128);
        layernorm128<<<N_NODES, blk128, 0, stream>>>(tmp, lngi, lnbi, xs);
        gemm_f32_wmma<1, 128><<<dim3(N_NODES / 16, 128 / 64), blk128, 0, stream>>>(
            o_t, opwi, opbi, tmp, N_NODES, 128);
        layernorm128<<<N_NODES, blk128, 0, stream>>>(tmp, lngi, lnbi, xt);
    }

    // ---- gated softmax pooling + classifier ----
    gate_kernel<<<N_NODES, blk128, 0, stream>>>(xs, Wg, bg, gat);
    pool_kernel<<<B_GRAPH, blk128, 0, stream>>>(xs, gat, pooled, 0);
    gate_kernel<<<N_NODES, blk128, 0, stream>>>(xt, Wg, bg, gat);
    pool_kernel<<<B_GRAPH, blk128, 0, stream>>>(xt, gat, pooled, 128);
    classify<<<1, 32, 0, stream>>>(pooled, Wc, bc, (float*)d_out);
}